// SparseLinear_32779190403590
// MI455X (gfx1250) — compile-verified
//
#include <hip/hip_runtime.h>
#include <stdint.h>

// Problem constants (match reference)
#define BATCH_N   4096
#define LENGTH_L  200
#define NCLASS    128
#define WPB       4                      // waves per block (wave32)
#define CHUNK     8                      // rows gathered per pipeline step
#define NCHUNK    (LENGTH_L / CHUNK)     // 25

#define WAIT_ASYNC(n) asm volatile("s_wait_asynccnt " #n ::: "memory")
#define WAIT_DS0()    asm volatile("s_wait_dscnt 0x0" ::: "memory")

// Issue one 512B embedding-row gather: 32 lanes x b128, async into this wave's LDS slot.
__device__ __forceinline__ void issue_row_async(const float* __restrict__ weight,
                                                int idx, int lane, const float4* slot) {
    const float4* g = reinterpret_cast<const float4*>(weight)
                    + (size_t)(unsigned)idx * (NCLASS / 4) + lane;
    // low 32 bits of a generic pointer into LDS == workgroup-relative LDS byte offset
    unsigned lds = (unsigned)(uintptr_t)slot;
    asm volatile("global_load_async_to_lds_b128 %0, %1, off"
                 :: "v"(lds), "v"(g)
                 : "memory");
}

__global__ __launch_bounds__(WPB * 32) void
sparse_linear_gather_kernel(const int* __restrict__ x,
                            const float* __restrict__ w,
                            const float* __restrict__ weight,
                            const float* __restrict__ bias,
                            float* __restrict__ out) {
    // [wave][buf(2)][row(CHUNK)][lane] float4 : 4 * 2 * 8 * 32 * 16B = 32 KB
    __shared__ float4 smem[WPB * 2 * CHUNK * 32];

    const int lane = threadIdx.x & 31;
    const int wv   = threadIdx.x >> 5;
    // force wave-uniform batch row -> index/weight fetches become scalar (SMEM) loads
    const int b = __builtin_amdgcn_readfirstlane((int)(blockIdx.x * WPB + wv));

    const int*   xb = x + (size_t)b * LENGTH_L;
    const float* wb = w + (size_t)b * LENGTH_L;
    float4* mybuf = &smem[(wv * 2 * CHUNK) * 32 + lane];   // slot(buf,i) = mybuf + (buf*CHUNK+i)*32

    // init accumulator with bias (out = sum + bias); lane owns channels [4*lane, 4*lane+4)
    float4 acc = *reinterpret_cast<const float4*>(bias + lane * 4);

    // ---- prologue: batch-load chunk 0 indices, issue chunk 0 into buf 0 ----
    {
        int idx0[CHUNK];
        #pragma unroll
        for (int i = 0; i < CHUNK; ++i) idx0[i] = xb[i];       // merged scalar loads
        #pragma unroll
        for (int i = 0; i < CHUNK; ++i)
            issue_row_async(weight, idx0[i], lane, mybuf + i * 32);
    }

    // ---- main loop: per chunk c -> prefetch scalars, issue chunk c+1, wait, consume c ----
    #pragma unroll 2
    for (int c = 0; c < NCHUNK - 1; ++c) {
        float wcur[CHUNK];
        int   idxn[CHUNK];
        #pragma unroll
        for (int i = 0; i < CHUNK; ++i) wcur[i] = wb[c * CHUNK + i];        // merged s_loads,
        #pragma unroll
        for (int i = 0; i < CHUNK; ++i) idxn[i] = xb[(c + 1) * CHUNK + i];  // one kmcnt wait

        // ensure prev chunk's ds reads of the buffer we're about to overwrite are retired
        WAIT_DS0();
        {
            float4* nb = mybuf + (((c + 1) & 1) * CHUNK) * 32;
            #pragma unroll
            for (int i = 0; i < CHUNK; ++i)
                issue_row_async(weight, idxn[i], lane, nb + i * 32);
        }

        // 16 rows outstanding; in-order completion => wait<=8 means chunk c is resident
        WAIT_ASYNC(8);
        {
            const float4* cb = mybuf + ((c & 1) * CHUNK) * 32;
            #pragma unroll
            for (int i = 0; i < CHUNK; ++i) {
                float4 v = cb[i * 32];                 // ds_load_b128
                acc.x = fmaf(wcur[i], v.x, acc.x);
                acc.y = fmaf(wcur[i], v.y, acc.y);
                acc.z = fmaf(wcur[i], v.z, acc.z);
                acc.w = fmaf(wcur[i], v.w, acc.w);
            }
        }
    }

    // ---- epilogue: drain and consume last chunk (c = NCHUNK-1, lives in buf 0) ----
    {
        float wcur[CHUNK];
        #pragma unroll
        for (int i = 0; i < CHUNK; ++i) wcur[i] = wb[(NCHUNK - 1) * CHUNK + i];
        WAIT_ASYNC(0);
        const float4* cb = mybuf + (((NCHUNK - 1) & 1) * CHUNK) * 32;
        #pragma unroll
        for (int i = 0; i < CHUNK; ++i) {
            float4 v = cb[i * 32];
            acc.x = fmaf(wcur[i], v.x, acc.x);
            acc.y = fmaf(wcur[i], v.y, acc.y);
            acc.z = fmaf(wcur[i], v.z, acc.z);
            acc.w = fmaf(wcur[i], v.w, acc.w);
        }
    }

    // coalesced 512B store of this batch row
    reinterpret_cast<float4*>(out + (size_t)b * NCLASS)[lane] = acc;
}

extern "C" void kernel_launch(void* const* d_in, const int* in_sizes, int n_in,
                              void* d_out, int out_size, void* d_ws, size_t ws_size,
                              hipStream_t stream) {
    (void)in_sizes; (void)n_in; (void)out_size; (void)d_ws; (void)ws_size;
    const int*   x      = (const int*)d_in[0];
    const float* w      = (const float*)d_in[1];
    const float* weight = (const float*)d_in[2];
    const float* bias   = (const float*)d_in[3];
    float*       out    = (float*)d_out;

    dim3 grid(BATCH_N / WPB);      // 1024 blocks
    dim3 block(WPB * 32);          // 128 threads = 4 wave32 waves
    hipLaunchKernelGGL(sparse_linear_gather_kernel, grid, block, 0, stream,
                       x, w, weight, bias, out);
}